// PointPooling_46677704573556
// MI455X (gfx1250) — compile-verified
//
#include <hip/hip_runtime.h>
#include <float.h>

// ---------------------------------------------------------------------------
// PointPooling on MI455X (gfx1250), fused single-kernel design, v2.
//
// score(n,m) = ||p_n||^2 - 2 * <s_m, p_n>   (same ranking as ||s_m - p_n||^2)
// Computed entirely inside V_WMMA_F32_16X16X4_F32 by packing:
//   A row n  = ( x_n, y_n, z_n, ||p_n||^2 )      [16x4 f32 A-matrix]
//   B col m  = (-2sx, -2sy, -2sz, 1.0 )          [4x16 f32 B-matrix]
// so D = A x B + 0 is directly the 16(n) x 16(m) score tile.
//
// v2 changes (from round-1 asm inspection):
//  - tile-min pre-filter: 7 v_min + 1 branch replaces 8 divergent checks
//    in the (common) no-insert case.
//  - branch-free A/B operand construction (v_cndmask instead of saveexec).
//  - 2-tile unroll: both WMMAs issued before divergent insert processing.
//  - merge rewritten: 128-thread parallel per-list sorts + cheap 8-way
//    sorted merge (take first 32), replacing the 16-thread O(32*256) scan.
// ---------------------------------------------------------------------------

typedef float v2f __attribute__((ext_vector_type(2)));
typedef float v8f __attribute__((ext_vector_type(8)));

constexpr int Bn = 4;       // batch
constexpr int Nn = 16384;   // points
constexpr int Mn = 4096;    // sample points
constexpr int Cn = 64;      // channels
constexpr int Kn = 32;      // poolN
constexpr int MT = 16;      // m-tile per workgroup
constexpr int WAVES = 4;    // waves per workgroup (wave32)
constexpr int NT = Nn / WAVES;
constexpr int PAD = 33;     // candidate stride (odd -> conflict-free banks)

__device__ __forceinline__ int part_base(int j, int p) {
    // source buffer for m-column j, part p (p = wave*2 + lane-half)
    return ((p >> 1) * 32 + j + (p & 1) * 16) * PAD;
}

__global__ __launch_bounds__(WAVES * 32)
void pointpool_kernel(const float* __restrict__ x,
                      const float* __restrict__ xyz,
                      const float* __restrict__ samp,
                      float* __restrict__ out) {
    __shared__ float cand_d[WAVES * 32 * PAD];
    __shared__ int   cand_i[WAVES * 32 * PAD];
    __shared__ int   sel_i[MT * PAD];   // final 32 neighbor ids per m
    __shared__ int   hd[MT * 8];        // merge head positions

    const int tid  = threadIdx.x;
    const int wave = tid >> 5;
    const int lane = tid & 31;
    const int half = lane >> 4;
    const int l15  = lane & 15;

    const int b  = blockIdx.x / (Mn / MT);
    const int m0 = (blockIdx.x % (Mn / MT)) * MT;

    // ---- B operand (fixed for whole workgroup), branch-free ----
    const float* srow = samp + ((size_t)b * Mn + (m0 + l15)) * 3;
    const float sx = srow[0], sy = srow[1], sz = srow[2];
    v2f Bop;
    Bop[0] = half ? (-2.0f * sz) : (-2.0f * sx);
    Bop[1] = half ? 1.0f         : (-2.0f * sy);

    // ---- init per-lane candidate buffer ----
    const int cbase = (wave * 32 + lane) * PAD;
#pragma unroll
    for (int k = 0; k < Kn; ++k) { cand_d[cbase + k] = FLT_MAX; cand_i[cbase + k] = 0; }
    float worst = FLT_MAX;
    int   wslot = 0;

    // ---- main loop: two 16-point tiles per iteration ----
    const float* xyzb   = xyz + (size_t)b * Nn * 3;
    const int    nbegin = wave * NT;
    const int    nend   = nbegin + NT;

    for (int n0 = nbegin; n0 < nend; n0 += 32) {
        const float* p0 = xyzb + (size_t)(n0 + l15) * 3;
        const float* p1 = xyzb + (size_t)(n0 + 16 + l15) * 3;
        const float ax0 = p0[0], ay0 = p0[1], az0 = p0[2];
        const float ax1 = p1[0], ay1 = p1[1], az1 = p1[2];
        const float nr0 = ax0 * ax0 + ay0 * ay0 + az0 * az0;
        const float nr1 = ax1 * ax1 + ay1 * ay1 + az1 * az1;

        v2f A0, A1;                       // branch-free operand build
        A0[0] = half ? az0 : ax0;  A0[1] = half ? nr0 : ay0;
        A1[0] = half ? az1 : ax1;  A1[1] = half ? nr1 : ay1;

        v8f Cz = {0.f, 0.f, 0.f, 0.f, 0.f, 0.f, 0.f, 0.f};
        v8f D0 = __builtin_amdgcn_wmma_f32_16x16x4_f32(
            false, A0, false, Bop, (short)0, Cz, false, false);
        v8f D1 = __builtin_amdgcn_wmma_f32_16x16x4_f32(
            false, A1, false, Bop, (short)0, Cz, false, false);

#pragma unroll
        for (int t = 0; t < 2; ++t) {
            const v8f D     = t ? D1 : D0;
            const int nrow0 = n0 + t * 16 + half * 8;

            // tile-level pre-filter: skip all 8 checks when nothing beats worst
            float tmin = D[0];
#pragma unroll
            for (int r = 1; r < 8; ++r) tmin = fminf(tmin, D[r]);

            if (tmin < worst) {
#pragma unroll
                for (int r = 0; r < 8; ++r) {
                    const float s = D[r];
                    if (s < worst) {
                        cand_d[cbase + wslot] = s;
                        cand_i[cbase + wslot] = nrow0 + r;
                        float wmax = -FLT_MAX; int wm = 0;
#pragma unroll
                        for (int k = 0; k < Kn; ++k) {
                            const float v = cand_d[cbase + k];
                            if (v > wmax) { wmax = v; wm = k; }
                        }
                        worst = wmax; wslot = wm;
                    }
                }
            }
        }
    }
    __syncthreads();

    // ---- stage 1: 128 threads sort their own 32-entry list (ascending) ----
    {
        const int j    = tid & 15;
        const int p    = tid >> 4;              // 0..7
        const int base = part_base(j, p);       // == own cbase, conflict-free
        for (int k = 0; k < Kn - 1; ++k) {
            int   best = k;
            float bd   = cand_d[base + k];
            for (int t = k + 1; t < Kn; ++t) {
                const float td = cand_d[base + t];
                if (td < bd) { bd = td; best = t; }
            }
            if (best != k) {
                const int pk = base + k, pb = base + best;
                const float td = cand_d[pk]; cand_d[pk] = cand_d[pb]; cand_d[pb] = td;
                const int   ti = cand_i[pk]; cand_i[pk] = cand_i[pb]; cand_i[pb] = ti;
            }
        }
    }
    __syncthreads();

    // ---- stage 2: 16 threads, 8-way merge of sorted lists, keep first 32 --
    if (tid < MT) {
        const int j = tid;
#pragma unroll
        for (int p = 0; p < 8; ++p) hd[j * 8 + p] = 0;
        for (int k = 0; k < Kn; ++k) {
            float bv = FLT_MAX; int bp = 0;
#pragma unroll
            for (int p = 0; p < 8; ++p) {
                const int pos = hd[j * 8 + p];
                const float v = (pos < Kn) ? cand_d[part_base(j, p) + pos] : FLT_MAX;
                if (v < bv) { bv = v; bp = p; }
            }
            const int pos = hd[j * 8 + bp];
            sel_i[j * PAD + k] = cand_i[part_base(j, bp) + pos];
            hd[j * 8 + bp] = pos + 1;
        }
    }
    __syncthreads();

    // ---- gather + max-pool: 8 threads per m, 8 channels each (float4 x2) --
    {
        const int j  = tid >> 3;
        const int c0 = (tid & 7) * 8;
        const float* xb = x + (size_t)b * Nn * Cn;
        float4 a0 = make_float4(-FLT_MAX, -FLT_MAX, -FLT_MAX, -FLT_MAX);
        float4 a1 = a0;
#pragma unroll 4
        for (int k = 0; k < Kn; ++k) {
            const int n = sel_i[j * PAD + k];               // broadcast read
            const float4* row = (const float4*)(xb + (size_t)n * Cn + c0);
            const float4 v0 = row[0], v1 = row[1];
            a0.x = fmaxf(a0.x, v0.x); a0.y = fmaxf(a0.y, v0.y);
            a0.z = fmaxf(a0.z, v0.z); a0.w = fmaxf(a0.w, v0.w);
            a1.x = fmaxf(a1.x, v1.x); a1.y = fmaxf(a1.y, v1.y);
            a1.z = fmaxf(a1.z, v1.z); a1.w = fmaxf(a1.w, v1.w);
        }
        float* orow = out + ((size_t)b * Mn + (m0 + j)) * Cn + c0;
        ((float4*)orow)[0] = a0;
        ((float4*)orow)[1] = a1;
    }
}

extern "C" void kernel_launch(void* const* d_in, const int* in_sizes, int n_in,
                              void* d_out, int out_size, void* d_ws, size_t ws_size,
                              hipStream_t stream) {
    const float* x    = (const float*)d_in[0];   // [B,N,C] f32
    const float* xyz  = (const float*)d_in[1];   // [B,N,3] f32
    const float* samp = (const float*)d_in[2];   // [B,M,3] f32
    float* out = (float*)d_out;                  // [B,M,C] f32

    dim3 grid(Bn * (Mn / MT));   // 1024 workgroups
    dim3 block(WAVES * 32);      // 4 wave32 waves
    hipLaunchKernelGGL(pointpool_kernel, grid, block, 0, stream,
                       x, xyz, samp, out);
}